// QDenseDirectedReupload_78142634983551
// MI455X (gfx1250) — compile-verified
//
#include <hip/hip_runtime.h>
#include <math.h>

// ---------------------------------------------------------------------------
// 11-qubit batched state-vector sim for MI455X (gfx1250).
//  - 1 workgroup (256 thr = 8 wave32) per sample; state in LDS (16 KB, in-place).
//  - Per layer: 3 fused 16x16-complex group unitaries via V_WMMA_F32_16X16X4_F32
//    chains (4 K-chunks x 4 real products = 16 WMMA/tile, 1 tile/wave).
//  - The 11-CNOT ring (GF(2)-linear index map) is absorbed into the *gather*
//    addressing of the next layer's first pass: zero extra LDS traffic.
//    A block-wide barrier between fragment-load and store makes the permuted
//    in-place pass race-free (all reads in registers before any write).
//  - Group matrices (shared across batch) precomputed into d_ws by qprep.
// ---------------------------------------------------------------------------

#define WIRES 11
#define DIM   2048
#define PIX   1024
#define NB    1024

typedef __attribute__((ext_vector_type(2))) float v2f;
typedef __attribute__((ext_vector_type(8))) float v8f;

// ---------------- precompute fused group matrices --------------------------
// G layout: [8 layers][3 groups][ re[16][16] | im[16][16] ]  (floats, 48 KB)
__global__ void qprep(const float* __restrict__ w0, const float* __restrict__ w1,
                      float* __restrict__ G) {
  const float PI_ = 3.14159265358979323846f;
  for (int e = threadIdx.x; e < 8 * 3 * 256; e += blockDim.x) {
    int L    = e / 768;
    int rem  = e - L * 768;
    int grp  = rem >> 8;
    int cell = rem & 255;
    int m = cell >> 4, k = cell & 15;
    const float* w = (L < 4) ? w0 : w1;
    int l = L & 3;

    float re = 1.f, im = 0.f;
    int nw, q0, topbit;
    bool zero = false;
    if (grp == 0)      { nw = 4; q0 = 0; topbit = 3; }
    else if (grp == 1) { nw = 4; q0 = 4; topbit = 3; }
    else               { nw = 3; q0 = 8; topbit = 2;      // I2 (x) G8 on bit3
                         if ((m >> 3) != (k >> 3)) zero = true; }
    if (zero) { re = 0.f; im = 0.f; }
    else {
      for (int i = 0; i < nw; ++i) {
        int q  = q0 + i;
        int bp = topbit - i;               // wire q0 sits at subspace MSB
        int mb = (m >> bp) & 1, kb = (k >> bp) & 1;
        const float* a3 = w + (l * WIRES + q) * 3;
        float phi   = PI_ * tanhf(a3[0]);   // qw_map.tanh
        float theta = PI_ * tanhf(a3[1]);
        float omega = PI_ * tanhf(a3[2]);
        float c = cosf(0.5f * theta), s = sinf(0.5f * theta);
        float h = 0.5f * (phi + omega), g = 0.5f * (phi - omega);
        float ur, ui;
        if (mb == 0 && kb == 0)      { ur =  cosf(h) * c; ui = -sinf(h) * c; } //  a*c
        else if (mb == 0 && kb == 1) { ur = -cosf(g) * s; ui = -sinf(g) * s; } // -b*s
        else if (mb == 1 && kb == 0) { ur =  cosf(g) * s; ui = -sinf(g) * s; } // conj(b)*s
        else                         { ur =  cosf(h) * c; ui =  sinf(h) * c; } // conj(a)*c
        float nr = re * ur - im * ui;
        float ni = re * ui + im * ur;
        re = nr; im = ni;
      }
    }
    int base = (L * 3 + grp) * 512;
    G[base + cell]       = re;
    G[base + 256 + cell] = im;
  }
}

// ---------------- inverse of the layer's CNOT-ring permutation -------------
// Forward: for q=0..10: CNOT(q, (q+r)%11) applied sequentially.
// Inverse = same CNOTs in reverse order (each CNOT is an involution).
// Wire q lives at index bit (10-q).
__device__ __forceinline__ int pinv_idx(int x, int r) {
#pragma unroll
  for (int q = WIRES - 1; q >= 0; --q) {
    int tq = q + r; if (tq >= WIRES) tq -= WIRES;
    x ^= ((x >> (10 - q)) & 1) << (10 - tq);
  }
  return x;
}

// ---------------- fused 16x16-complex group gate via f32 WMMA --------------
// Subspace bits [lo+3 : lo]; 128 cosets; wave handles cosets [wave*16, +16).
// D(16x16) = G(16x16,complex) * S(16x16,complex) per tile:
//   Dr = Gr*Sr + (-Gi)*Si ; Di = Gr*Si + Gi*Sr   (4 K-chunks each -> 16 WMMA)
// rperm != 0: gather reads through the pending CNOT-ring inverse permutation;
// a block barrier between load and store makes the in-place update race-free.
__device__ __forceinline__ void apply_group(float2* st,
                                            const float* __restrict__ gr,
                                            const float* __restrict__ gi,
                                            int lo, int wave, int lane, int rperm) {
  const int n    = lane & 15;               // tile column (coset) / A row
  const int half = lane >> 4;
  const int j    = wave * 16 + n;           // global coset id 0..127
  const int base = ((j >> lo) << (lo + 4)) | (j & ((1 << lo) - 1));

  v2f ar[4], ai[4], br[4], bi[4];
#pragma unroll
  for (int c = 0; c < 4; ++c) {
    int k0 = 4 * c + 2 * half;              // K rows held by this lane half
    int i0 = base | (k0 << lo);
    int i1 = base | ((k0 + 1) << lo);
    if (rperm) { i0 = pinv_idx(i0, rperm); i1 = pinv_idx(i1, rperm); }
    float2 s0 = st[i0];
    float2 s1 = st[i1];
    br[c] = v2f{s0.x, s1.x};
    bi[c] = v2f{s0.y, s1.y};
    ar[c] = v2f{gr[n * 16 + k0], gr[n * 16 + k0 + 1]};
    ai[c] = v2f{gi[n * 16 + k0], gi[n * 16 + k0 + 1]};
  }
  if (rperm) __syncthreads();               // whole state now in registers block-wide

  v8f dr = {0, 0, 0, 0, 0, 0, 0, 0};
  v8f di = {0, 0, 0, 0, 0, 0, 0, 0};
#pragma unroll
  for (int c = 0; c < 4; ++c) {
    v2f nai = -ai[c];                       // f32 WMMA has no A-negate; do it in VALU
    dr = __builtin_amdgcn_wmma_f32_16x16x4_f32(false, ar[c], false, br[c],
                                               (short)0, dr, false, false);
    dr = __builtin_amdgcn_wmma_f32_16x16x4_f32(false, nai,   false, bi[c],
                                               (short)0, dr, false, false);
    di = __builtin_amdgcn_wmma_f32_16x16x4_f32(false, ar[c], false, bi[c],
                                               (short)0, di, false, false);
    di = __builtin_amdgcn_wmma_f32_16x16x4_f32(false, ai[c], false, br[c],
                                               (short)0, di, false, false);
  }

#pragma unroll
  for (int r = 0; r < 8; ++r) {             // D: VGPR r = rows r / r+8
    int m = r + 8 * half;
    st[base | (m << lo)] = make_float2(dr[r], di[r]);  // canonical layout
  }
}

// ---------------- main simulation kernel -----------------------------------
__global__ void __launch_bounds__(256)
qsim(const float* __restrict__ x, const float* __restrict__ y,
     const float* __restrict__ G, float* __restrict__ out) {
  __shared__ float2 st[DIM];                // single in-place state, 16 KB
  __shared__ float  red[256];
  const int tid  = threadIdx.x;
  const int lane = tid & 31;
  const int wave = tid >> 5;
  const int b    = blockIdx.x;

  // ---- amplitude embedding (normalize=True); pixel p -> amplitude 2p ----
  float xs[4], part = 0.f;
#pragma unroll
  for (int t = 0; t < 4; ++t) {
    int p = tid + 256 * t;
    float v = x[b * PIX + p];
    xs[t] = v; part += v * v;
  }
  red[tid] = part; __syncthreads();
  for (int off = 128; off > 0; off >>= 1) {
    if (tid < off) red[tid] += red[tid + off];
    __syncthreads();
  }
  float inv = rsqrtf(red[0]);
#pragma unroll
  for (int t = 0; t < 4; ++t) {
    int p = tid + 256 * t;
    st[2 * p]     = make_float2(xs[t] * inv, 0.f);
    st[2 * p + 1] = make_float2(0.f, 0.f);
  }
  __syncthreads();

  const float yv = y[b];
  const float rc = cosf(0.5f * yv), rs = sinf(0.5f * yv);

  int pend = 0;                             // pending (un-materialized) CNOT range
  for (int blkw = 0; blkw < 2; ++blkw) {
    // ---- per-sample RX(y) on wire 10, addressed through pending perm ----
    // logical pair (2p, 2p+1) lives at {Pinv(2p), Pinv(2p+1)}; pairs stay
    // disjoint per thread (Pinv is a bijection), so in-place is safe.
#pragma unroll
    for (int t = 0; t < 4; ++t) {
      int p  = tid + 256 * t;
      int x0 = 2 * p, x1 = 2 * p + 1;
      if (pend) { x0 = pinv_idx(x0, pend); x1 = pinv_idx(x1, pend); }
      float2 a0 = st[x0], a1 = st[x1];
      st[x0] = make_float2(rc * a0.x + rs * a1.y, rc * a0.y - rs * a1.x);
      st[x1] = make_float2(rc * a1.x + rs * a0.y, rc * a1.y - rs * a0.x);
    }
    __syncthreads();

    for (int l = 0; l < 4; ++l) {
      int L = blkw * 4 + l;
      const float* g0 = G + (L * 3 + 0) * 512;   // wires 0-3  (bits 10..7)
      apply_group(st, g0, g0 + 256, 7, wave, lane, pend); __syncthreads();
      const float* g1 = G + (L * 3 + 1) * 512;   // wires 4-7  (bits 6..3)
      apply_group(st, g1, g1 + 256, 3, wave, lane, 0);    __syncthreads();
      const float* g2 = G + (L * 3 + 2) * 512;   // wires 8-10 (bits 3..0, I(x)G8)
      apply_group(st, g2, g2 + 256, 0, wave, lane, 0);    __syncthreads();
      pend = l + 1;                              // this layer's CNOT ring, deferred
    }
  }

  // ---- probs of even basis states (through final pending perm, r=4) ----
#pragma unroll
  for (int t = 0; t < 4; ++t) {
    int p  = tid + 256 * t;
    float2 a = st[pinv_idx(2 * p, pend)];
    out[b * PIX + p] = fminf(1.0f, (a.x * a.x + a.y * a.y) * (float)PIX);
  }
}

// ---------------------------------------------------------------------------
extern "C" void kernel_launch(void* const* d_in, const int* in_sizes, int n_in,
                              void* d_out, int out_size, void* d_ws, size_t ws_size,
                              hipStream_t stream) {
  const float* x  = (const float*)d_in[0];   // (B,1,32,32) f32
  const float* y  = (const float*)d_in[1];   // (B,) f32
  const float* w0 = (const float*)d_in[2];   // (4,11,3) f32
  const float* w1 = (const float*)d_in[3];   // (4,11,3) f32
  float* G = (float*)d_ws;                   // 8*3*512 floats = 48 KB scratch

  qprep<<<1, 256, 0, stream>>>(w0, w1, G);
  qsim<<<NB, 256, 0, stream>>>(x, y, G, (float*)d_out);
}